// WDLoss_39341900431444
// MI455X (gfx1250) — compile-verified
//
#include <hip/hip_runtime.h>

#define TILE         4096
#define SORT_THREADS  256
#define ITEMS          16   // TILE / SORT_THREADS
#define SCAN_CHUNKS   256
#define DIFF_BLOCKS  1024
#define DIFF_THREADS  256

typedef float v2f __attribute__((ext_vector_type(2)));
typedef float v8f __attribute__((ext_vector_type(8)));

// ---------- order-preserving float <-> uint key ----------
__device__ __forceinline__ unsigned f2k(float f) {
    unsigned b = __float_as_uint(f);
    unsigned mask = ((unsigned)(((int)b) >> 31)) | 0x80000000u; // neg:0xFFFFFFFF pos:0x80000000
    return b ^ mask;
}
__device__ __forceinline__ float k2f(unsigned k) {
    unsigned mask = (k & 0x80000000u) ? 0x80000000u : 0xFFFFFFFFu;
    return __uint_as_float(k ^ mask);
}

// ---------- wave32 scans ----------
__device__ __forceinline__ unsigned waveInclScan(unsigned x) {
    int lane = (int)(threadIdx.x & 31u);
    #pragma unroll
    for (int off = 1; off < 32; off <<= 1) {
        unsigned y = __shfl_up(x, off);
        if (lane >= off) x += y;
    }
    return x;
}

// Block exclusive scan over blockDim.x = NWAVES*32 threads.
// waveBuf must have NWAVES+1 entries of LDS.
template <int NWAVES>
__device__ __forceinline__ unsigned blockExclScan(unsigned v, unsigned* waveBuf,
                                                  unsigned& total) {
    unsigned lane = threadIdx.x & 31u;
    unsigned wid  = threadIdx.x >> 5;
    unsigned incl = waveInclScan(v);
    if (lane == 31) waveBuf[wid] = incl;
    __syncthreads();
    if (wid == 0) {
        unsigned s  = (lane < NWAVES) ? waveBuf[lane] : 0u;
        unsigned si = waveInclScan(s);
        if (lane < NWAVES) waveBuf[lane] = si - s;        // exclusive wave offsets
        if (lane == NWAVES - 1) waveBuf[NWAVES] = si;     // grand total
    }
    __syncthreads();
    total = waveBuf[NWAVES];
    unsigned r = incl - v + waveBuf[wid];
    __syncthreads();   // protect waveBuf for next call
    return r;
}

// ---------- kernels ----------
// hist[d * numBlocks + blk] = count of digit d in this block's tile.
// If fsrc != nullptr (pass 0), keys come from on-the-fly float->key conversion.
__global__ void hist_kernel(const unsigned* __restrict__ in, const float* __restrict__ fsrc,
                            unsigned* __restrict__ hist, int shift, int numBlocks) {
    __shared__ unsigned cnt[256];
    int t = threadIdx.x;
    cnt[t] = 0u;
    __syncthreads();
    size_t base = (size_t)blockIdx.x * TILE;
    if (fsrc) {
        #pragma unroll
        for (int j = 0; j < ITEMS; ++j) {
            unsigned k = f2k(fsrc[base + (size_t)j * SORT_THREADS + t]);
            atomicAdd(&cnt[(k >> shift) & 255u], 1u);
        }
    } else {
        #pragma unroll
        for (int j = 0; j < ITEMS; ++j) {
            unsigned k = in[base + (size_t)j * SORT_THREADS + t];
            atomicAdd(&cnt[(k >> shift) & 255u], 1u);
        }
    }
    __syncthreads();
    hist[(size_t)t * numBlocks + blockIdx.x] = cnt[t];
}

// Hierarchical exclusive scan, stage 1: each block scans one contiguous chunk
// (chunkLen entries, multiple of 1024) in place and records its chunk total.
__global__ void scan_chunk_kernel(unsigned* __restrict__ data,
                                  unsigned* __restrict__ chunkTotals, int chunkLen) {
    __shared__ unsigned waveBuf[33];
    unsigned carry = 0u;
    int t = threadIdx.x;
    size_t base0 = (size_t)blockIdx.x * chunkLen;
    for (int base = 0; base < chunkLen; base += 1024) {
        unsigned v = data[base0 + base + t];
        unsigned total;
        unsigned excl = blockExclScan<32>(v, waveBuf, total);
        data[base0 + base + t] = excl + carry;
        carry += total;   // uniform across threads
    }
    if (t == 0) chunkTotals[blockIdx.x] = carry;
}

// Stage 2: exclusive scan of the chunk totals (n <= 256), single block.
__global__ void scan_totals_kernel(unsigned* __restrict__ totals, int n) {
    __shared__ unsigned waveBuf[9];
    int t = threadIdx.x;
    unsigned v = (t < n) ? totals[t] : 0u;
    unsigned total;
    unsigned excl = blockExclScan<8>(v, waveBuf, total);
    if (t < n) totals[t] = excl;
}

// Stage 3: add per-chunk offsets.
__global__ void add_offsets_kernel(unsigned* __restrict__ data,
                                   const unsigned* __restrict__ totals, int chunkLen) {
    unsigned off = totals[blockIdx.x];
    size_t base0 = (size_t)blockIdx.x * chunkLen;
    for (int i = threadIdx.x; i < chunkLen; i += blockDim.x)
        data[base0 + i] += off;
}

// Stable scatter: local 8x bit-split counting sort in LDS, then write to
// hist-provided global offsets using digit-boundary positions.
// If fsrc != nullptr (pass 0), keys come from on-the-fly float->key conversion.
__global__ void scatter_kernel(const unsigned* __restrict__ in, const float* __restrict__ fsrc,
                               unsigned* __restrict__ out,
                               const unsigned* __restrict__ hist, int shift, int numBlocks) {
    __shared__ unsigned bufA[TILE];
    __shared__ unsigned bufB[TILE];
    __shared__ unsigned waveBuf[9];
    __shared__ unsigned digitStart[256];

    int t = threadIdx.x;
    size_t base = (size_t)blockIdx.x * TILE;

    // coalesced load into LDS (convert on the fly for pass 0)
    if (fsrc) {
        #pragma unroll
        for (int j = 0; j < ITEMS; ++j) {
            int idx = j * SORT_THREADS + t;
            bufA[idx] = f2k(fsrc[base + idx]);
        }
    } else {
        #pragma unroll
        for (int j = 0; j < ITEMS; ++j) {
            int idx = j * SORT_THREADS + t;
            bufA[idx] = in[base + idx];
        }
    }
    __syncthreads();

    // 8 stable bit-splits: bufA -> bufB -> bufA -> ... ends in bufA
    #pragma unroll
    for (int bit = 0; bit < 8; ++bit) {
        unsigned* src = (bit & 1) ? bufB : bufA;
        unsigned* dst = (bit & 1) ? bufA : bufB;
        int sh = shift + bit;

        unsigned cnt0 = 0;
        #pragma unroll
        for (int j = 0; j < ITEMS; ++j) {
            unsigned k = src[t * ITEMS + j];
            cnt0 += (((k >> sh) & 1u) == 0u);
        }
        unsigned total0;
        unsigned z = blockExclScan<8>(cnt0, waveBuf, total0); // zeros before my items

        #pragma unroll
        for (int j = 0; j < ITEMS; ++j) {
            int idx = t * ITEMS + j;
            unsigned k = src[idx];
            if (((k >> sh) & 1u) == 0u) {
                dst[z] = k;
                ++z;                      // z == zeros among tile elements < idx, +1
            } else {
                dst[total0 + (unsigned)idx - z] = k; // ones-before = idx - zeros-before
            }
        }
        __syncthreads();
    }

    // digit boundary detection on locally sorted tile
    for (int idx = t; idx < TILE; idx += SORT_THREADS) {
        unsigned d = (bufA[idx] >> shift) & 255u;
        if (idx == 0 || d != ((bufA[idx - 1] >> shift) & 255u))
            digitStart[d] = (unsigned)idx;
    }
    __syncthreads();

    // scatter to global (runs within a digit are contiguous)
    for (int idx = t; idx < TILE; idx += SORT_THREADS) {
        unsigned k = bufA[idx];
        unsigned d = (k >> shift) & 255u;
        unsigned pos = hist[(size_t)d * numBlocks + blockIdx.x] + (unsigned)idx - digitStart[d];
        out[pos] = k;
    }
}

// Streaming |u_(i) - v_(i)| with WMMA-based in-register reduction.
// Each lane feeds 2 abs-diffs per iteration as the f32 A-matrix (16x4),
// B = ones, C accumulates. Every A element appears in 16 columns of D,
// so Sum(all D) / 16 == Sum of fed values.
__global__ void diff_wmma_kernel(const unsigned* __restrict__ ku,
                                 const unsigned* __restrict__ kv,
                                 float* __restrict__ partials, int npairs) {
    const uint2* __restrict__ ku2 = (const uint2*)ku;
    const uint2* __restrict__ kv2 = (const uint2*)kv;

    v8f acc = {};
    v2f bones;
    bones.x = 1.0f; bones.y = 1.0f;

    int stride = gridDim.x * blockDim.x;
    for (int p = blockIdx.x * blockDim.x + threadIdx.x; p < npairs; p += stride) {
        __builtin_prefetch(&ku2[p + stride], 0, 0);   // global_prefetch_b8 (speculative, safe OOB)
        __builtin_prefetch(&kv2[p + stride], 0, 0);
        uint2 a = ku2[p];
        uint2 b = kv2[p];
        v2f tt;
        tt.x = fabsf(k2f(a.x) - k2f(b.x));
        tt.y = fabsf(k2f(a.y) - k2f(b.y));
        // D = A x B + C : v_wmma_f32_16x16x4_f32
        acc = __builtin_amdgcn_wmma_f32_16x16x4_f32(
            /*neg_a=*/false, tt, /*neg_b=*/false, bones,
            /*c_mod=*/(short)0, acc, /*reuse_a=*/false, /*reuse_b=*/false);
    }

    float s = acc[0] + acc[1] + acc[2] + acc[3] + acc[4] + acc[5] + acc[6] + acc[7];
    #pragma unroll
    for (int m = 16; m >= 1; m >>= 1) s += __shfl_xor(s, m);

    __shared__ float wsum[DIFF_THREADS / 32];
    if ((threadIdx.x & 31u) == 0u) wsum[threadIdx.x >> 5] = s;
    __syncthreads();
    if (threadIdx.x == 0) {
        float bsum = 0.0f;
        #pragma unroll
        for (int i = 0; i < DIFF_THREADS / 32; ++i) bsum += wsum[i];
        partials[blockIdx.x] = bsum * (1.0f / 16.0f);
    }
}

__global__ void final_reduce_kernel(const float* __restrict__ partials, int n,
                                    float* __restrict__ out, float invN) {
    __shared__ float sh[256];
    int t = threadIdx.x;
    float s = 0.0f;
    for (int i = t; i < n; i += 256) s += partials[i];   // fixed order -> deterministic
    sh[t] = s;
    __syncthreads();
    for (int off = 128; off > 0; off >>= 1) {
        if (t < off) sh[t] += sh[t + off];
        __syncthreads();
    }
    if (t == 0) out[0] = sh[0] * invN;
}

// ---------- host ----------
extern "C" void kernel_launch(void* const* d_in, const int* in_sizes, int n_in,
                              void* d_out, int out_size, void* d_ws, size_t ws_size,
                              hipStream_t stream) {
    if (n_in < 2) return;
    const float* u = (const float*)d_in[0];
    const float* v = (const float*)d_in[1];
    float* out = (float*)d_out;

    const int N = in_sizes[0];                 // 16,777,216
    const int numBlocks = N / TILE;            // 4096  (N divisible by TILE)
    const int histEntries = 256 * numBlocks;   // 1,048,576
    const int chunkLen = histEntries / SCAN_CHUNKS; // 4096 (multiple of 1024)
    const int npairs = N / 2;

    // workspace layout (~272 MB): two ping-pong buffers per array + hist + scan/reduce temps
    unsigned* kU0 = (unsigned*)d_ws;                                  // N
    unsigned* kU1 = kU0 + (size_t)N;                                  // N
    unsigned* kV0 = kU1 + (size_t)N;                                  // N
    unsigned* kV1 = kV0 + (size_t)N;                                  // N
    unsigned* hist = kV1 + (size_t)N;                                 // histEntries
    unsigned* chunkTotals = hist + (size_t)histEntries;               // SCAN_CHUNKS
    float* partials = (float*)(chunkTotals + SCAN_CHUNKS);            // DIFF_BLOCKS

    auto scanHist = [&]() {
        scan_chunk_kernel<<<SCAN_CHUNKS, 1024, 0, stream>>>(hist, chunkTotals, chunkLen);
        scan_totals_kernel<<<1, 256, 0, stream>>>(chunkTotals, SCAN_CHUNKS);
        add_offsets_kernel<<<SCAN_CHUNKS, 256, 0, stream>>>(hist, chunkTotals, chunkLen);
    };

    // 4-pass LSD radix sort (8-bit digits), stable.
    // Pass 0 converts float->key on the fly (saves a full makekeys round trip).
    // dst chain: fsrc -> b0 -> b1 -> b0 -> b1 ; final sorted keys live in b1.
    auto sort4 = [&](const float* fsrc, unsigned* b0, unsigned* b1) -> unsigned* {
        // pass 0: from floats
        hist_kernel<<<numBlocks, SORT_THREADS, 0, stream>>>(nullptr, fsrc, hist, 0, numBlocks);
        scanHist();
        scatter_kernel<<<numBlocks, SORT_THREADS, 0, stream>>>(nullptr, fsrc, b0, hist, 0, numBlocks);
        // passes 1..3: key buffers
        unsigned* src = b0;
        unsigned* dst = b1;
        for (int p = 1; p < 4; ++p) {
            int shift = p * 8;
            hist_kernel<<<numBlocks, SORT_THREADS, 0, stream>>>(src, nullptr, hist, shift, numBlocks);
            scanHist();
            scatter_kernel<<<numBlocks, SORT_THREADS, 0, stream>>>(src, nullptr, dst, hist, shift, numBlocks);
            unsigned* tmp = src; src = dst; dst = tmp;
        }
        return src;   // after 3 swaps this is b1
    };

    unsigned* sortedU = sort4(u, kU0, kU1);
    unsigned* sortedV = sort4(v, kV0, kV1);

    // mean |u_(i) - v_(i)| via WMMA reduction, deterministic two-stage finish
    diff_wmma_kernel<<<DIFF_BLOCKS, DIFF_THREADS, 0, stream>>>(sortedU, sortedV, partials, npairs);
    final_reduce_kernel<<<1, 256, 0, stream>>>(partials, DIFF_BLOCKS, out, 1.0f / (float)N);
    (void)out_size; (void)ws_size;
}